// ChebyshevHelmholtzSolver2D_25924422599050
// MI455X (gfx1250) — compile-verified
//
#include <hip/hip_runtime.h>

#define N_TOT 4096
#define NB 64

typedef __attribute__((ext_vector_type(2))) float v2f;
typedef __attribute__((ext_vector_type(4))) float v4f;
typedef __attribute__((ext_vector_type(8))) float v8f;

// ---------------------------------------------------------------------------
// Assemble A (row-major 4096x4096) and RHS b in workspace.
// r = ry*64+rx (row-major vec of (y,x)); boundary rows -> identity, b -> bc.
// ---------------------------------------------------------------------------
__global__ void assemble_kernel(const float* __restrict__ f,
                                const float* __restrict__ alpha,
                                const float* __restrict__ bc_top,
                                const float* __restrict__ bc_bottom,
                                const float* __restrict__ bc_left,
                                const float* __restrict__ bc_right,
                                const float* __restrict__ Dx2,
                                const float* __restrict__ Dy2,
                                float* __restrict__ A,
                                float* __restrict__ b)
{
    int idx = blockIdx.x * blockDim.x + threadIdx.x;   // exactly n*n threads
    int r = idx >> 12;
    int c = idx & (N_TOT - 1);
    int ry = r >> 6, rx = r & 63;
    int cy = c >> 6, cx = c & 63;
    bool bnd = (ry == 0) || (ry == 63) || (rx == 0) || (rx == 63);
    float val;
    if (bnd) {
        val = (r == c) ? 1.0f : 0.0f;
    } else {
        val = 0.0f;
        if (ry == cy) val += Dx2[(rx << 6) + cx];   // kron(Iy, Dx2)
        if (rx == cx) val += Dy2[(ry << 6) + cy];   // kron(Dy2, Ix)
        if (r == c)  val -= alpha[0];
    }
    A[idx] = val;
    if (c == 0) {
        float bv;
        if (!bnd)           bv = f[r];
        else if (rx == 0)   bv = bc_right[ry];   // col 0 <- bc_right (incl corners)
        else if (rx == 63)  bv = bc_left[ry];    // col 63 <- bc_left (incl corners)
        else if (ry == 0)   bv = bc_top[rx];
        else                bv = bc_bottom[rx];
        b[r] = bv;
    }
}

// ---------------------------------------------------------------------------
// Unblocked panel factorization: columns [p, p+NB), rows [p, n). No pivoting.
// One 1024-thread workgroup; pivot row segment staged in LDS.
// ---------------------------------------------------------------------------
__global__ void lu_panel_kernel(float* __restrict__ A, int p)
{
    __shared__ float prow[NB];
    const int tid = threadIdx.x;
    for (int jj = 0; jj < NB; ++jj) {
        const int j = p + jj;
        const int w = NB - jj;                 // pivot col .. end of panel
        if (tid < w) prow[tid] = A[j * N_TOT + j + tid];
        __syncthreads();
        const float pivinv = 1.0f / prow[0];
        for (int i = j + 1 + tid; i < N_TOT; i += 1024) {
            float* Ai = A + i * N_TOT;
            float l = Ai[j] * pivinv;
            Ai[j] = l;
            for (int k = 1; k < w; ++k) Ai[j + k] -= l * prow[k];
        }
        __syncthreads();
    }
}

// ---------------------------------------------------------------------------
// TRSM: U12 = L11^{-1} * A12. L11 (unit lower, 64x64) staged in LDS,
// one thread per column of A12, forward substitution in registers.
// ---------------------------------------------------------------------------
__global__ void lu_trsm_kernel(float* __restrict__ A, int p, int rem)
{
    __shared__ float L[NB][NB];
    for (int t = threadIdx.x; t < NB * NB; t += blockDim.x)
        L[t >> 6][t & 63] = A[(p + (t >> 6)) * N_TOT + p + (t & 63)];
    __syncthreads();
    const int c = blockIdx.x * blockDim.x + threadIdx.x;
    if (c >= rem) return;
    float* col = A + p * N_TOT + (p + NB) + c;
    float x[NB];
    for (int i = 0; i < NB; ++i) {
        float a = col[i * N_TOT];
        for (int j = 0; j < i; ++j) a -= L[i][j] * x[j];
        x[i] = a;
        col[i * N_TOT] = a;
    }
}

// ---------------------------------------------------------------------------
// Trailing update: C -= L21 * U12 with V_WMMA_F32_16X16X4_F32.
// Block = 8 waves covering 128(M) x 64(N) of C. The 64x64 U12 block is staged
// once per block into LDS *transposed* (Bt[col][k], pitch 66 -> the (k,k+1)
// pair per lane is one aligned 8-byte LDS read; transpose writes avoid
// 32-way bank conflicts). Each wave computes a 16x64 strip (4 accumulators),
// A operand reused across the 4 WMMAs per K-step; K = 64, step 4.
//
// RMW elimination: accumulators are initialized to -C before the K loop, WMMA
// accumulates L*U on top, and the epilogue stores -acc = C - L*U. This turns
// the write-back into a pure store clause (no serialized load/sub/store).
// ---------------------------------------------------------------------------
__global__ __launch_bounds__(256) void lu_gemm_kernel(float* __restrict__ A,
                                                      int p, int rem, int Mb)
{
    __shared__ float Bt[64][66];             // Bt[col][k] = U12[k][col]
    const int bm = blockIdx.x % Mb;
    const int bs = blockIdx.x / Mb;
    const int base = p + NB;
    const int colBase = base + (bs << 6);

    // Cooperative stage: coalesced 128-bit row loads, transposed LDS writes.
    for (int t = threadIdx.x; t < 1024; t += 256) {
        const int kRow = t >> 4;             // 0..63
        const int col  = (t & 15) << 2;      // 0,4,...,60
        const v4f vb = *(const v4f*)(A + (p + kRow) * N_TOT + colBase + col);
        Bt[col + 0][kRow] = vb.x;
        Bt[col + 1][kRow] = vb.y;
        Bt[col + 2][kRow] = vb.z;
        Bt[col + 3][kRow] = vb.w;
    }
    __syncthreads();

    const int wib  = threadIdx.x >> 5;
    const int lane = threadIdx.x & 31;
    const int rowTile = (bm << 7) + (wib << 4);
    if (rowTile >= rem) return;              // wave-uniform guard
    const int rowBase = base + rowTile;
    const int hi = lane >> 4;
    const int lo = lane & 15;

    // Load C tile negated into the accumulators (coalesced load clauses).
    v8f acc0, acc1, acc2, acc3;
#pragma unroll
    for (int v = 0; v < 8; ++v) {
        const float* Crow = A + (rowBase + v + (hi << 3)) * N_TOT + colBase + lo;
        acc0[v] = -Crow[0];
        acc1[v] = -Crow[16];
        acc2[v] = -Crow[32];
        acc3[v] = -Crow[48];
    }

    const float* Arow = A + (rowBase + lo) * N_TOT + p;   // L21 row
#pragma unroll 4
    for (int k = 0; k < NB; k += 4) {
        const int kk = k + (hi << 1);
        // A 16x4: lane(hi,lo) holds A[lo][kk], A[lo][kk+1]  (8B aligned)
        v2f a = *(const v2f*)(Arow + kk);
        // B 4x16: lane(hi,lo) holds B[kk][col], B[kk+1][col] from LDS
        v2f bv0 = *(const v2f*)(&Bt[lo][kk]);
        v2f bv1 = *(const v2f*)(&Bt[lo + 16][kk]);
        v2f bv2 = *(const v2f*)(&Bt[lo + 32][kk]);
        v2f bv3 = *(const v2f*)(&Bt[lo + 48][kk]);
        acc0 = __builtin_amdgcn_wmma_f32_16x16x4_f32(false, a, false, bv0, (short)0, acc0, false, false);
        acc1 = __builtin_amdgcn_wmma_f32_16x16x4_f32(false, a, false, bv1, (short)0, acc1, false, false);
        acc2 = __builtin_amdgcn_wmma_f32_16x16x4_f32(false, a, false, bv2, (short)0, acc2, false, false);
        acc3 = __builtin_amdgcn_wmma_f32_16x16x4_f32(false, a, false, bv3, (short)0, acc3, false, false);
    }
    // acc = L*U - C_old ; store C_new = -acc. Pure store clause, no RMW chain.
    // C/D layout: VGPR v, lane(hi,lo): row = v + 8*hi, col = lo
#pragma unroll
    for (int v = 0; v < 8; ++v) {
        float* Crow = A + (rowBase + v + (hi << 3)) * N_TOT + colBase + lo;
        Crow[0]  = -acc0[v];
        Crow[16] = -acc1[v];
        Crow[32] = -acc2[v];
        Crow[48] = -acc3[v];
    }
}

// ---------------------------------------------------------------------------
// Forward substitution  L y = b (unit lower). Single 1024-thread block,
// solution staged in LDS; each thread owns 4 rows.
// ---------------------------------------------------------------------------
__global__ void fwd_kernel(const float* __restrict__ A,
                           const float* __restrict__ b,
                           float* __restrict__ y)
{
    __shared__ float ys[N_TOT];
    const int tid = threadIdx.x;
    float s0 = b[tid], s1 = b[tid + 1024], s2 = b[tid + 2048], s3 = b[tid + 3072];
    for (int j = 0; j < N_TOT; ++j) {
        if ((j & 1023) == tid) {
            const int q = j >> 10;
            const float v = (q == 0) ? s0 : (q == 1) ? s1 : (q == 2) ? s2 : s3;
            ys[j] = v;
            y[j] = v;
        }
        __syncthreads();
        const float yj = ys[j];
        if (tid        > j) s0 -= A[tid          * N_TOT + j] * yj;
        if (tid + 1024 > j) s1 -= A[(tid + 1024) * N_TOT + j] * yj;
        if (tid + 2048 > j) s2 -= A[(tid + 2048) * N_TOT + j] * yj;
        if (tid + 3072 > j) s3 -= A[(tid + 3072) * N_TOT + j] * yj;
    }
}

// ---------------------------------------------------------------------------
// Backward substitution  U x = y. Writes final solution to d_out.
// ---------------------------------------------------------------------------
__global__ void bwd_kernel(const float* __restrict__ A,
                           const float* __restrict__ y,
                           float* __restrict__ x)
{
    __shared__ float xs[N_TOT];
    const int tid = threadIdx.x;
    float s0 = y[tid], s1 = y[tid + 1024], s2 = y[tid + 2048], s3 = y[tid + 3072];
    for (int j = N_TOT - 1; j >= 0; --j) {
        if ((j & 1023) == tid) {
            const int q = j >> 10;
            float v = (q == 0) ? s0 : (q == 1) ? s1 : (q == 2) ? s2 : s3;
            v /= A[j * N_TOT + j];
            xs[j] = v;
            x[j] = v;
        }
        __syncthreads();
        const float xj = xs[j];
        if (tid        < j) s0 -= A[tid          * N_TOT + j] * xj;
        if (tid + 1024 < j) s1 -= A[(tid + 1024) * N_TOT + j] * xj;
        if (tid + 2048 < j) s2 -= A[(tid + 2048) * N_TOT + j] * xj;
        if (tid + 3072 < j) s3 -= A[(tid + 3072) * N_TOT + j] * xj;
    }
}

extern "C" void kernel_launch(void* const* d_in, const int* in_sizes, int n_in,
                              void* d_out, int out_size, void* d_ws, size_t ws_size,
                              hipStream_t stream)
{
    (void)in_sizes; (void)n_in; (void)out_size;
    const float* f   = (const float*)d_in[0];
    const float* al  = (const float*)d_in[1];
    const float* bt  = (const float*)d_in[2];
    const float* bb  = (const float*)d_in[3];
    const float* bl  = (const float*)d_in[4];
    const float* br  = (const float*)d_in[5];
    const float* Dx2 = (const float*)d_in[6];
    const float* Dy2 = (const float*)d_in[7];
    float* out = (float*)d_out;

    const size_t need = ((size_t)N_TOT * N_TOT + 2u * N_TOT) * sizeof(float);
    if (ws_size < need) return;   // cannot run without A-matrix workspace

    float* A = (float*)d_ws;
    float* b = A + (size_t)N_TOT * N_TOT;
    float* y = b + N_TOT;

    assemble_kernel<<<(N_TOT * N_TOT) / 256, 256, 0, stream>>>(
        f, al, bt, bb, bl, br, Dx2, Dy2, A, b);

    for (int p = 0; p < N_TOT; p += NB) {
        lu_panel_kernel<<<1, 1024, 0, stream>>>(A, p);
        const int rem = N_TOT - p - NB;
        if (rem > 0) {
            lu_trsm_kernel<<<(rem + 255) / 256, 256, 0, stream>>>(A, p, rem);
            const int Mb = (rem + 127) >> 7;          // 128-row blocks
            const int blocks = Mb * (rem >> 6);       // x 64-col strips
            lu_gemm_kernel<<<blocks, 256, 0, stream>>>(A, p, rem, Mb);
        }
    }

    fwd_kernel<<<1, 1024, 0, stream>>>(A, b, y);
    bwd_kernel<<<1, 1024, 0, stream>>>(A, y, out);
}